// ChebGCN_4844723109937
// MI455X (gfx1250) — compile-verified
//
#include <hip/hip_runtime.h>
#include <hip/hip_bf16.h>

// Problem constants (match reference)
#define NN   50000
#define EE   800000
#define FF   64
#define GG   64
#define CC   10

typedef __attribute__((ext_vector_type(2))) float v2f;
typedef __attribute__((ext_vector_type(8))) float v8f;
typedef __attribute__((ext_vector_type(4))) int   v4i;

#if defined(__AMDGCN__) && __has_builtin(__builtin_amdgcn_global_load_async_to_lds_b128) && __has_builtin(__builtin_amdgcn_s_wait_asynccnt)
#define USE_ASYNC_LDS 1
#else
#define USE_ASYNC_LDS 0
#endif

// ---------------- elementwise / setup kernels ----------------

__global__ void zero_kernel(float* __restrict__ p, int n) {
    int i = blockIdx.x * blockDim.x + threadIdx.x;
    if (i < n) p[i] = 0.0f;
}

__global__ void negcopy_kernel(float* __restrict__ dst, const float* __restrict__ src, int n) {
    int i = blockIdx.x * blockDim.x + threadIdx.x;
    if (i < n) dst[i] = -src[i];
}

__global__ void deg_scatter_kernel(float* __restrict__ deg,
                                   const float* __restrict__ w,
                                   const int* __restrict__ row, int e_cnt) {
    int e = blockIdx.x * blockDim.x + threadIdx.x;
    if (e < e_cnt) atomicAdd(&deg[row[e]], w[e]);
}

__global__ void dinv_kernel(float* __restrict__ deg, int n) {
    int i = blockIdx.x * blockDim.x + threadIdx.x;
    if (i < n) {
        float d = deg[i];
        deg[i] = (d > 0.0f) ? rsqrtf(d) : 0.0f;
    }
}

__global__ void norm_kernel(float* __restrict__ norm,
                            const float* __restrict__ w,
                            const float* __restrict__ dinv,
                            const int* __restrict__ row,
                            const int* __restrict__ col, int e_cnt) {
    int e = blockIdx.x * blockDim.x + threadIdx.x;
    if (e < e_cnt) norm[e] = -w[e] * dinv[row[e]] * dinv[col[e]];
}

// ---------------- sparse propagation: dst[row] += scale*norm[e]*src[col] ----------------
// 16 lanes per edge, float4 per lane -> one global_load_b128 per lane, coalesced
// 256B gather per edge; 4 L2 float atomics per lane.

__global__ void prop_scatter_kernel(float* __restrict__ dst,
                                    const float* __restrict__ src,
                                    const float* __restrict__ norm,
                                    const int* __restrict__ row,
                                    const int* __restrict__ col,
                                    float scale, int e_cnt) {
    long long idx = (long long)blockIdx.x * blockDim.x + threadIdx.x;
    int e  = (int)(idx >> 4);
    int f4 = (int)(idx & 15);
    if (e >= e_cnt) return;
    float wv = scale * norm[e];
    const float4* s = (const float4*)(src + (size_t)col[e] * FF);
    float4 v = s[f4];
    float* d = dst + (size_t)row[e] * FF + f4 * 4;
    atomicAdd(d,     wv * v.x);
    atomicAdd(d + 1, wv * v.y);
    atomicAdd(d + 2, wv * v.z);
    atomicAdd(d + 3, wv * v.w);
}

// ---------------- fused Cheb output: out = T0*W0 + T1*W1 + T2*W2 + b (opt ReLU) --------
// One block = 16 output rows, 4 waves; wave w owns output columns [16w,16w+16).
// fp32 WMMA 16x16x4: A lanes 0-15 -> K={k0,k0+1}, lanes 16-31 -> K={k0+2,k0+3};
// C/D: vgpr r, lane L -> (M = r + 8*(L>>4), N = L&15).
// A-tile (16x64 = 4KB) is staged to LDS with async B128 DMA when available.

__launch_bounds__(128)
__global__ void cheb_wmma_kernel(const float* __restrict__ T0,
                                 const float* __restrict__ T1,
                                 const float* __restrict__ T2,
                                 const float* __restrict__ W,    // [3][64][64]
                                 const float* __restrict__ bias, // [64]
                                 float* __restrict__ out,
                                 int relu) {
    __shared__ float tileA[16 * FF];
    const int tid   = threadIdx.x;
    const int wave  = tid >> 5;
    const int lane  = tid & 31;
    const int l16   = lane & 15;
    const int lhalf = lane >> 4;
    const int rowBase = blockIdx.x * 16;          // N divisible by 16
    const int ncol  = wave * 16 + l16;            // global output column

    v8f c;
    float bv = bias[ncol];
    #pragma unroll
    for (int r = 0; r < 8; ++r) c[r] = bv;

    const float* terms[3] = {T0, T1, T2};
    for (int t = 0; t < 3; ++t) {
        const float* src = terms[t] + (size_t)rowBase * FF;
        __syncthreads();
#if USE_ASYNC_LDS
        {
            // 16*64 floats = 4KB = 256 x 16B granules; 128 threads x 2 granules
            #pragma unroll
            for (int i = 0; i < 2; ++i) {
                int o = (tid + i * 128) * 4;      // float index of this 16B granule
                __builtin_amdgcn_global_load_async_to_lds_b128(
                    (__attribute__((address_space(1))) v4i*)(src + o),
                    (__attribute__((address_space(3))) v4i*)(tileA + o),
                    0, 0);
            }
            __builtin_amdgcn_s_wait_asynccnt(0);
        }
#else
        {
            const float4* s4 = (const float4*)src;
            float4* d4 = (float4*)tileA;
            for (int i = tid; i < 16 * FF / 4; i += 128) d4[i] = s4[i];
        }
#endif
        __syncthreads();
        const float* Wt = W + t * FF * FF;
        #pragma unroll
        for (int k0 = 0; k0 < FF; k0 += 4) {
            int ka = k0 + 2 * lhalf;
            v2f a, b;
            a.x = tileA[l16 * FF + ka];
            a.y = tileA[l16 * FF + ka + 1];
            b.x = Wt[(size_t)ka * FF + ncol];
            b.y = Wt[(size_t)(ka + 1) * FF + ncol];
            c = __builtin_amdgcn_wmma_f32_16x16x4_f32(
                    false, a, false, b, (short)0, c, false, false);
        }
    }

    #pragma unroll
    for (int r = 0; r < 8; ++r) {
        float v = c[r];
        if (relu) v = v > 0.0f ? v : 0.0f;
        out[(size_t)(rowBase + r + 8 * lhalf) * FF + ncol] = v;
    }
}

// ---------------- pooling ----------------
// 16 lanes per node, float4 per lane.

__global__ void pool_kernel(float* __restrict__ pooled, float* __restrict__ cnt,
                            const float* __restrict__ h,
                            const int* __restrict__ batch, int n_cnt) {
    long long idx = (long long)blockIdx.x * blockDim.x + threadIdx.x;
    int i  = (int)(idx >> 4);
    int f4 = (int)(idx & 15);
    if (i >= n_cnt) return;
    int g = batch[i];
    const float4* s = (const float4*)(h + (size_t)i * FF);
    float4 v = s[f4];
    float* d = pooled + (size_t)g * FF + f4 * 4;
    atomicAdd(d,     v.x);
    atomicAdd(d + 1, v.y);
    atomicAdd(d + 2, v.z);
    atomicAdd(d + 3, v.w);
    if (f4 == 0) atomicAdd(&cnt[g], 1.0f);
}

// ---------------- final tiny MLP: [64,64] -> relu[64,32] -> [64,10] ----------------

__launch_bounds__(256)
__global__ void mlp_kernel(const float* __restrict__ pooled, const float* __restrict__ cnt,
                           const float* __restrict__ Wl1, const float* __restrict__ bl1,
                           const float* __restrict__ Wl2, const float* __restrict__ bl2,
                           float* __restrict__ out) {
    __shared__ float pm[GG * FF];
    __shared__ float hid[GG * 32];
    int tid = threadIdx.x;
    for (int i = tid; i < GG * FF; i += 256) {
        int g = i >> 6;
        float cv = cnt[g];
        cv = cv > 1.0f ? cv : 1.0f;
        pm[i] = pooled[i] / cv;
    }
    __syncthreads();
    for (int i = tid; i < GG * 32; i += 256) {
        int g = i >> 5, j = i & 31;
        float acc = bl1[j];
        #pragma unroll 8
        for (int f = 0; f < FF; ++f) acc += pm[g * FF + f] * Wl1[f * 32 + j];
        hid[i] = acc > 0.0f ? acc : 0.0f;
    }
    __syncthreads();
    for (int i = tid; i < GG * CC; i += 256) {
        int g = i / CC, cc = i % CC;
        float acc = bl2[cc];
        #pragma unroll
        for (int j = 0; j < 32; ++j) acc += hid[g * 32 + j] * Wl2[j * CC + cc];
        out[i] = acc;
    }
}

// ---------------- host driver ----------------

extern "C" void kernel_launch(void* const* d_in, const int* in_sizes, int n_in,
                              void* d_out, int out_size, void* d_ws, size_t ws_size,
                              hipStream_t stream) {
    const float* x    = (const float*)d_in[0];
    const int*   ei   = (const int*)d_in[1];
    const float* ea   = (const float*)d_in[2];
    const int*   batch= (const int*)d_in[3];
    const float* W1   = (const float*)d_in[4];
    const float* b1   = (const float*)d_in[5];
    const float* W2   = (const float*)d_in[6];
    const float* b2   = (const float*)d_in[7];
    const float* W3   = (const float*)d_in[8];
    const float* b3   = (const float*)d_in[9];
    const float* Wl1  = (const float*)d_in[10];
    const float* bl1  = (const float*)d_in[11];
    const float* Wl2  = (const float*)d_in[12];
    const float* bl2  = (const float*)d_in[13];

    const int* row = ei;        // edge_index[0]
    const int* col = ei + EE;   // edge_index[1]

    float* ws     = (float*)d_ws;
    float* deg    = ws;                       // N
    float* norm_  = deg + NN;                 // E
    float* Tx1    = norm_ + EE;               // N*64
    float* Tx2    = Tx1 + (size_t)NN * FF;    // N*64
    float* hA     = Tx2 + (size_t)NN * FF;    // N*64
    float* hB     = hA  + (size_t)NN * FF;    // N*64
    float* pooled = hB  + (size_t)NN * FF;    // G*64
    float* cnt    = pooled + GG * FF;         // G

    const int TB = 256;
    // degree + laplacian edge norms
    zero_kernel<<<(NN + TB - 1) / TB, TB, 0, stream>>>(deg, NN);
    deg_scatter_kernel<<<(EE + TB - 1) / TB, TB, 0, stream>>>(deg, ea, row, EE);
    dinv_kernel<<<(NN + TB - 1) / TB, TB, 0, stream>>>(deg, NN);
    norm_kernel<<<(EE + TB - 1) / TB, TB, 0, stream>>>(norm_, ea, deg, row, col, EE);

    const long long propThreads = (long long)EE * 16;
    const int propBlocks = (int)((propThreads + TB - 1) / TB);
    const int nf = NN * FF;
    const int nfBlocks = (nf + TB - 1) / TB;

    auto run_layer = [&](const float* X, const float* W, const float* b,
                         float* outp, int relu) {
        zero_kernel<<<nfBlocks, TB, 0, stream>>>(Tx1, nf);
        prop_scatter_kernel<<<propBlocks, TB, 0, stream>>>(Tx1, X, norm_, row, col, 1.0f, EE);
        negcopy_kernel<<<nfBlocks, TB, 0, stream>>>(Tx2, X, nf);                 // Tx2 = -T0
        prop_scatter_kernel<<<propBlocks, TB, 0, stream>>>(Tx2, Tx1, norm_, row, col, 2.0f, EE);
        cheb_wmma_kernel<<<NN / 16, 128, 0, stream>>>(X, Tx1, Tx2, W, b, outp, relu);
    };

    run_layer(x,  W1, b1, hA, 1);
    run_layer(hA, W2, b2, hB, 1);
    run_layer(hB, W3, b3, hA, 0);

    zero_kernel<<<(GG * FF + GG + TB - 1) / TB, TB, 0, stream>>>(pooled, GG * FF + GG);
    pool_kernel<<<(int)(((long long)NN * 16 + TB - 1) / TB), TB, 0, stream>>>(pooled, cnt, hA, batch, NN);
    mlp_kernel<<<1, 256, 0, stream>>>(pooled, cnt, Wl1, bl1, Wl2, bl2, (float*)d_out);
}